// Quantize_29085518528713
// MI455X (gfx1250) — compile-verified
//
#include <hip/hip_runtime.h>

// ---------------------------------------------------------------------------
// VQ-VAE quantize for MI455X (gfx1250, wave32).
// Fused f32-WMMA GEMM + argmin (V_WMMA_F32_16X16X4_F32), EMA updates.
// Round 2: packed WMMA-ready B (b128 loads), 2 row tiles per wave (B reuse).
// ---------------------------------------------------------------------------

typedef __attribute__((ext_vector_type(2))) float v2f;
typedef __attribute__((ext_vector_type(4))) float v4f;
typedef __attribute__((ext_vector_type(8))) float v8f;

#define DECAY_F      0.99f
#define OMD_F        0.01f      // 1 - decay
#define EPS_F        1e-5f

#define BATCH  16
#define DIM    64
#define HH     64
#define WW     64
#define NEMB   2048
#define NTOT   (BATCH * HH * WW)          // 65536 flat vectors
#define XELEMS (BATCH * DIM * HH * WW)    // 4194304

// ---------------------------------------------------------------------------
// Kernel 1: hnorm[e] = 0.5 * ||embed[:,e]||^2
// ---------------------------------------------------------------------------
__global__ __launch_bounds__(256) void k_hnorm(const float* __restrict__ embed,
                                               float* __restrict__ hnorm) {
    const int e = blockIdx.x * 256 + threadIdx.x;
    float s = 0.0f;
#pragma unroll
    for (int k = 0; k < DIM; ++k) {
        const float v = embed[k * NEMB + e];   // coalesced across lanes
        s += v * v;
    }
    hnorm[e] = 0.5f * s;
}

// ---------------------------------------------------------------------------
// Kernel 1b: repack embed into WMMA B-operand lane order so the main loop can
// use global_load_b128.  Layout: packB[t][p][lane] = 4 floats covering
// K-steps s=2p and s=2p+1 for this lane:
//   lane = hi*16 + m, col = t*16 + m
//   {embed[8p+2hi][col], embed[8p+2hi+1][col],
//    embed[8p+4+2hi][col], embed[8p+5+2hi][col]}
// Total 128*8*32*4 floats = 512 KB (one-time shuffle, L2-resident after).
// ---------------------------------------------------------------------------
__global__ __launch_bounds__(256) void k_packB(const float* __restrict__ embed,
                                               v4f* __restrict__ packB) {
    const int gid = blockIdx.x * 256 + threadIdx.x;  // 32768 groups
    const int l  = gid & 31;
    const int p  = (gid >> 5) & 7;
    const int t  = gid >> 8;
    const int hi = l >> 4, m = l & 15;
    const int col = t * 16 + m;
    const int k0  = 8 * p + 2 * hi;
    v4f o;
    o.x = embed[(k0    ) * NEMB + col];
    o.y = embed[(k0 + 1) * NEMB + col];
    o.z = embed[(k0 + 4) * NEMB + col];
    o.w = embed[(k0 + 5) * NEMB + col];
    packB[gid] = o;
}

// ---------------------------------------------------------------------------
// Kernel 2: fused  score = flat @ embed - 0.5||e||^2  with running argmax.
// One wave owns TWO 16-row tiles (32 rows) so every B register pair feeds two
// v_wmma_f32_16x16x4_f32.  Per column tile: 8 global_load_b128 + 32 WMMAs.
//
// A layout (32-bit A 16x4, ISA 7.12.2): lanes 0-15 = M=0..15 K={k0,k0+1},
// lanes 16-31 = M=0..15 K={k0+2,k0+3}.  C (16x16 f32): VGPR r -> rows r
// (lanes 0-15) / r+8 (lanes 16-31).
// ---------------------------------------------------------------------------
__global__ __launch_bounds__(256) void k_dist_argmin(
    const float* __restrict__ x, const v4f* __restrict__ packB,
    const float* __restrict__ hnorm, int* __restrict__ wind,
    float* __restrict__ out_ind) {
    const int lane = threadIdx.x & 31;
    const int wave = threadIdx.x >> 5;
    const int m    = lane & 15;
    const int hi   = lane >> 4;
    const int row_base = (blockIdx.x * 8 + wave) * 32;

    // Preload both 16x64 A panels in WMMA register layout (64 VGPRs).
    v2f a0[16], a1[16];
    {
        const int  row = row_base + m;
        const int  bb = row >> 12, hh2 = (row >> 6) & 63, w = row & 63;
        const long xb = (long)bb * (DIM * HH * WW) + hh2 * WW + w;
#pragma unroll
        for (int s = 0; s < 16; ++s) {
            const int d0 = 4 * s + 2 * hi;
            a0[s].x = x[xb + (long)d0 * (HH * WW)];
            a0[s].y = x[xb + (long)(d0 + 1) * (HH * WW)];
        }
    }
    {
        const int  row = row_base + 16 + m;
        const int  bb = row >> 12, hh2 = (row >> 6) & 63, w = row & 63;
        const long xb = (long)bb * (DIM * HH * WW) + hh2 * WW + w;
#pragma unroll
        for (int s = 0; s < 16; ++s) {
            const int d0 = 4 * s + 2 * hi;
            a1[s].x = x[xb + (long)d0 * (HH * WW)];
            a1[s].y = x[xb + (long)(d0 + 1) * (HH * WW)];
        }
    }

    float best0[8], best1[8];
    int   bidx0[8], bidx1[8];
#pragma unroll
    for (int r = 0; r < 8; ++r) {
        best0[r] = -3.4e38f; bidx0[r] = 0;
        best1[r] = -3.4e38f; bidx1[r] = 0;
    }

    for (int t = 0; t < NEMB / 16; ++t) {
        const int base = t * 256 + lane;           // 8 groups of 32 lanes
        __builtin_prefetch(&packB[base + 256], 0, 0);

        v2f  bv[16];
        v4f* bv4 = (v4f*)bv;
#pragma unroll
        for (int j = 0; j < 8; ++j) bv4[j] = packB[base + j * 32];

        v8f c0 = {}, c1 = {};
#pragma unroll
        for (int s = 0; s < 16; ++s) {
            c0 = __builtin_amdgcn_wmma_f32_16x16x4_f32(
                     false, a0[s], false, bv[s], (short)0, c0, false, false);
            c1 = __builtin_amdgcn_wmma_f32_16x16x4_f32(
                     false, a1[s], false, bv[s], (short)0, c1, false, false);
        }

        const int   col = t * 16 + m;
        const float hn  = hnorm[col];
#pragma unroll
        for (int r = 0; r < 8; ++r) {
            const float sc0 = c0[r] - hn;          // maximize f.e - 0.5||e||^2
            if (sc0 > best0[r]) { best0[r] = sc0; bidx0[r] = col; }
            const float sc1 = c1[r] - hn;
            if (sc1 > best1[r]) { best1[r] = sc1; bidx1[r] = col; }
        }
    }

    // Cross-lane argmax within each 16-lane half (rows r and r+8).
#pragma unroll
    for (int r = 0; r < 8; ++r) {
        float v  = best0[r];
        int   id = bidx0[r];
#pragma unroll
        for (int off = 8; off >= 1; off >>= 1) {
            const float ov  = __shfl_xor(v, off);
            const int   oid = __shfl_xor(id, off);
            if (ov > v || (ov == v && oid < id)) { v = ov; id = oid; }
        }
        if (m == 0) {
            const int ro = row_base + r + 8 * hi;
            wind[ro]    = id;
            out_ind[ro] = (float)id;
        }
    }
#pragma unroll
    for (int r = 0; r < 8; ++r) {
        float v  = best1[r];
        int   id = bidx1[r];
#pragma unroll
        for (int off = 8; off >= 1; off >>= 1) {
            const float ov  = __shfl_xor(v, off);
            const int   oid = __shfl_xor(id, off);
            if (ov > v || (ov == v && oid < id)) { v = ov; id = oid; }
        }
        if (m == 0) {
            const int ro = row_base + 16 + r + 8 * hi;
            wind[ro]    = id;
            out_ind[ro] = (float)id;
        }
    }
}

// ---------------------------------------------------------------------------
// Kernel 3: gather code -> out (NCHW, identical flat layout as x),
// diff partial sums, and segment sums (counts, flat.T @ onehot).
// ---------------------------------------------------------------------------
__global__ __launch_bounds__(256) void k_gather(
    const float* __restrict__ x, const float* __restrict__ embed,
    const int* __restrict__ wind, float* __restrict__ out,
    float* __restrict__ esum, float* __restrict__ counts,
    float* __restrict__ wdiff) {
    __shared__ float sdata[256];
    const int n = blockIdx.x * 256 + threadIdx.x;
    const int d = blockIdx.y;
    const int idx = wind[n];
    const int bb = n >> 12, hh2 = (n >> 6) & 63, w = n & 63;
    const long off = (long)bb * (DIM * HH * WW) + (long)d * (HH * WW) + hh2 * WW + w;

    const float xv = x[off];
    const float q  = embed[d * NEMB + idx];
    out[off] = q;

    atomicAdd(&esum[d * NEMB + idx], xv);
    if (d == 0) atomicAdd(&counts[idx], 1.0f);

    const float e = q - xv;
    sdata[threadIdx.x] = e * e;
    __syncthreads();
    for (int s = 128; s > 0; s >>= 1) {
        if (threadIdx.x < s) sdata[threadIdx.x] += sdata[threadIdx.x + s];
        __syncthreads();
    }
    if (threadIdx.x == 0) atomicAdd(wdiff, sdata[0]);
}

// ---------------------------------------------------------------------------
// Kernel 4a (single block): EMA cluster sizes + total n + diff finalize.
// ---------------------------------------------------------------------------
__global__ __launch_bounds__(256) void k_ema_cs(
    const float* __restrict__ cluster_size, const float* __restrict__ counts,
    float* __restrict__ out_ncs, float* __restrict__ wn,
    const float* __restrict__ wdiff, float* __restrict__ out_diff) {
    __shared__ float sdata[256];
    float loc = 0.0f;
    for (int e = threadIdx.x; e < NEMB; e += 256) {
        const float ncs = DECAY_F * cluster_size[e] + OMD_F * counts[e];
        out_ncs[e] = ncs;
        loc += ncs;
    }
    sdata[threadIdx.x] = loc;
    __syncthreads();
    for (int s = 128; s > 0; s >>= 1) {
        if (threadIdx.x < s) sdata[threadIdx.x] += sdata[threadIdx.x + s];
        __syncthreads();
    }
    if (threadIdx.x == 0) {
        wn[0] = sdata[0];
        out_diff[0] = wdiff[0] * (1.0f / (float)XELEMS);
    }
}

// ---------------------------------------------------------------------------
// Kernel 4b: EMA embed_avg + normalized new embed.
// ---------------------------------------------------------------------------
__global__ __launch_bounds__(256) void k_ema_embed(
    const float* __restrict__ embed_avg, const float* __restrict__ esum,
    const float* __restrict__ out_ncs, const float* __restrict__ wn,
    float* __restrict__ out_nea, float* __restrict__ out_ne) {
    const int g = blockIdx.x * 256 + threadIdx.x;   // 0 .. DIM*NEMB-1
    const int e = g & (NEMB - 1);
    const float nea = DECAY_F * embed_avg[g] + OMD_F * esum[g];
    out_nea[g] = nea;
    const float ntot = wn[0];
    const float ncs  = out_ncs[e];
    const float cs   = (ncs + EPS_F) / (ntot + (float)NEMB * EPS_F) * ntot;
    out_ne[g] = nea / cs;
}

// ---------------------------------------------------------------------------
// Launch
// ---------------------------------------------------------------------------
extern "C" void kernel_launch(void* const* d_in, const int* in_sizes, int n_in,
                              void* d_out, int out_size, void* d_ws, size_t ws_size,
                              hipStream_t stream) {
    const float* x            = (const float*)d_in[0];   // [16,64,64,64]
    const float* embed        = (const float*)d_in[1];   // [64,2048]
    const float* cluster_size = (const float*)d_in[2];   // [2048]
    const float* embed_avg    = (const float*)d_in[3];   // [64,2048]

    // d_out: out | diff | embed_ind | new_embed | new_cluster_size | new_embed_avg
    float* outp     = (float*)d_out;
    float* out_diff = outp + 4194304;
    float* out_ind  = outp + 4194305;
    float* out_ne   = outp + 4259841;
    float* out_ncs  = outp + 4390913;
    float* out_nea  = outp + 4392961;

    // Workspace layout (floats)
    float* ws_f   = (float*)d_ws;
    float* hnorm  = ws_f;                  // 2048
    float* counts = ws_f + 2048;           // 2048
    float* esum   = ws_f + 4096;           // 131072
    float* wdiff  = ws_f + 135168;         // 1
    float* wn     = ws_f + 135169;         // 1
    int*   wind   = (int*)(ws_f + 135170); // 65536 ints
    v4f*   packB  = (v4f*)(ws_f + 200708); // 131072 floats (16B aligned)

    // Zero the accumulators (counts + esum + diff + n) each call.
    hipMemsetAsync(counts, 0, (size_t)(2048 + 131072 + 2) * sizeof(float), stream);

    k_hnorm<<<NEMB / 256, 256, 0, stream>>>(embed, hnorm);
    k_packB<<<(128 * 8 * 32) / 256, 256, 0, stream>>>(embed, packB);

    // 2048 * 32-row wave tiles, 8 waves per 256-thread block -> 256 blocks.
    k_dist_argmin<<<NTOT / (32 * 8), 256, 0, stream>>>(x, packB, hnorm, wind, out_ind);

    dim3 g3(NTOT / 256, DIM);
    k_gather<<<g3, 256, 0, stream>>>(x, embed, wind, outp, esum, counts, wdiff);

    k_ema_cs<<<1, 256, 0, stream>>>(cluster_size, counts, out_ncs, wn, wdiff, out_diff);

    k_ema_embed<<<(DIM * NEMB) / 256, 256, 0, stream>>>(embed_avg, esum, out_ncs, wn,
                                                        out_nea, out_ne);
}